// DepthToPointCloud_37580963840692
// MI455X (gfx1250) — compile-verified
//
#include <hip/hip_runtime.h>
#include <stdint.h>

// ---------------- problem constants ----------------
#define H_IMG 1080
#define W_IMG 1920
#define N_PTS (H_IMG * W_IMG)      // 2073600
#define M_PTS 2048
#define FX_ 1050.0f
#define FY_ 1050.0f
#define CX_ 960.0f
#define CY_ 540.0f
#define MIN_D 0.1f
#define MAX_D 2.0f

// ---------------- launch geometry ----------------
#define NB 128                      // persistent blocks (must all be resident)
#define NT 1024                     // threads per block = 32 wave32 waves
#define TOT (NB * NT)               // 131072 threads
#define PPT 16                      // points per thread; NB*NT*PPT = 2097152 >= N_PTS
#define CHUNK (NT * PPT)            // 16384 points per block -> 64 KB LDS depth tile

#define WS_U64 (1 + M_PTS)          // [cnt,epoch] + argmax slots

typedef unsigned long long u64;
typedef unsigned int u32;

// ---------------- TDM availability / arity guard ----------------
#if defined(__gfx1250__) && __has_builtin(__builtin_amdgcn_tensor_load_to_lds)
#define HAVE_TDM 1
typedef unsigned int tdm_v4u __attribute__((ext_vector_type(4)));
typedef int tdm_v8i __attribute__((ext_vector_type(8)));
typedef int tdm_v4i __attribute__((ext_vector_type(4)));
#endif

// monotone unsigned key for float max (handles +/-inf correctly)
__device__ __forceinline__ u32 fkey(float f) {
  u32 b = __float_as_uint(f);
  return ((int)b < 0) ? ~b : (b | 0x80000000u);
}

__device__ __forceinline__ u64 wave_max_u64(u64 v) {
#pragma unroll
  for (int off = 16; off > 0; off >>= 1) {
    u64 o = __shfl_xor(v, off, 32);
    v = (o > v) ? o : v;
  }
  return v;
}

// device-wide barrier: epoch counter in global workspace, zeroed each call
__device__ __forceinline__ void grid_barrier(u32* cnt, u32* epoch, u32 target) {
  __syncthreads();
  if (threadIdx.x == 0) {
    u32 prev = __hip_atomic_fetch_add(cnt, 1u, __ATOMIC_ACQ_REL, __HIP_MEMORY_SCOPE_AGENT);
    if (prev == (u32)(NB - 1)) {
      __hip_atomic_store(cnt, 0u, __ATOMIC_RELAXED, __HIP_MEMORY_SCOPE_AGENT);
      __hip_atomic_fetch_add(epoch, 1u, __ATOMIC_RELEASE, __HIP_MEMORY_SCOPE_AGENT);
    } else {
      while (__hip_atomic_load(epoch, __ATOMIC_ACQUIRE, __HIP_MEMORY_SCOPE_AGENT) < target) {
        __builtin_amdgcn_s_sleep(2);
      }
    }
  }
  __syncthreads();
}

// recompute a point's xyz from its flat index (invalid -> +inf, like reference)
__device__ __forceinline__ void point_from_index(const float* __restrict__ depth,
                                                 int widx, float& px, float& py, float& pz) {
  float d = depth[widx];
  bool valid = (d > MIN_D) && (d < MAX_D);   // also excludes NaN/inf/<=0
  int u = widx % W_IMG;
  int v = widx / W_IMG;
  float inf = __builtin_inff();
  px = valid ? ((float)u - CX_) * d * (1.0f / FX_) : inf;
  py = valid ? ((float)v - CY_) * d * (1.0f / FY_) : inf;
  pz = valid ? d : inf;
}

__global__ void init_ws_kernel(u64* __restrict__ ws) {
  int i = blockIdx.x * blockDim.x + threadIdx.x;
  if (i < WS_U64) ws[i] = 0ull;
}

// ---------------- persistent FPS kernel ----------------
__global__ __launch_bounds__(NT) void fps_kernel(const float* __restrict__ depth,
                                                 u64* __restrict__ ws) {
  __shared__ float sdepth[CHUNK];          // 64 KB: block's depth tile (setup only)
  __shared__ u64 sred[NT / 32];            // 32 per-wave partials

  const int t = threadIdx.x;
  const int b = blockIdx.x;
  const int base = b * NT + t;             // point index of (row 0, this lane)

  // ---- stage depth tile: rows of NT elements, row stride = TOT elements ----
#ifdef HAVE_TDM
  if (t == 0) {
    uint64_t ga = (uint64_t)(uintptr_t)(depth + (size_t)b * NT);
    u32 lds_addr = (u32)(uintptr_t)(&sdepth[0]);
    u32 nrows = (u32)((N_PTS - 1 - b * NT) / TOT) + 1u;   // in-bounds rows; rest zero-fill
    // D# group 0: count=1 | lds_addr | global_addr[56:0] | type=2
    tdm_v4u g0 = { 1u,
                   lds_addr,
                   (u32)ga,
                   (u32)((ga >> 32) & 0x01FFFFFFu) | (2u << 30) };
    const u32 td0 = (u32)TOT;             // tensor_dim0 = 131072 elements
    const u64 str0 = (u64)TOT;            // tensor_dim0_stride = 131072 elements
    tdm_v8i g1 = {
      (int)(2u << 16),                                              // data_size=4B, mask=0
      (int)((td0 & 0xFFFFu) << 16),                                 // dim0[15:0] @ 63:48
      (int)(((td0 >> 16) & 0xFFFFu) | ((nrows & 0xFFFFu) << 16)),   // dim0[31:16] | dim1 lo
      (int)(((nrows >> 16) & 0xFFFFu) | ((u32)NT << 16)),           // dim1 hi | tile_dim0=1024
      (int)((u32)PPT & 0xFFFFu),                                    // tile_dim1=16, tile_dim2=0
      (int)(u32)(str0 & 0xFFFFFFFFull),                             // stride0 lo32
      (int)(u32)((str0 >> 32) & 0xFFFFull),                         // stride0 hi16 | stride1 lo=0
      0                                                             // stride1 hi
    };
    tdm_v4i gz = {0, 0, 0, 0};
#if __clang_major__ >= 23
    tdm_v8i gz8 = {0, 0, 0, 0, 0, 0, 0, 0};
    __builtin_amdgcn_tensor_load_to_lds(g0, g1, gz, gz, gz8, 0);
#else
    __builtin_amdgcn_tensor_load_to_lds(g0, g1, gz, gz, 0);
#endif
  }
#if __has_builtin(__builtin_amdgcn_s_wait_tensorcnt)
  __builtin_amdgcn_s_wait_tensorcnt(0);
#endif
  __syncthreads();
#else
#pragma unroll
  for (int i = 0; i < PPT; ++i) {
    int pidx = base + i * TOT;
    sdepth[i * NT + t] = (pidx < N_PTS) ? depth[pidx] : 0.0f;
  }
  __syncthreads();
#endif

  // ---- per-thread register state: loop-invariant xyz + running dists ----
  // 4*16 = 64 VGPRs of state -> stays well below the 256-VGPR direct window
  // (avoids S_SET_VGPR_MSB SALU overhead seen with 32 points/thread).
  float X[PPT], Y[PPT], Z[PPT], dist[PPT];
  const float inf = __builtin_inff();
  const float qnan = __builtin_nanf("");
#pragma unroll
  for (int i = 0; i < PPT; ++i) {
    int pidx = base + i * TOT;
    float d = sdepth[i * NT + t];
    bool inb = (pidx < N_PTS);
    bool valid = (d > MIN_D) && (d < MAX_D);
    int u = pidx % W_IMG;
    int v = pidx / W_IMG;
    // invalid points: NaN coords -> candidate NaN -> v_min_num ignores -> dist stays +inf
    X[i] = valid ? ((float)u - CX_) * (1.0f / FX_) * d : qnan;
    Y[i] = valid ? ((float)v - CY_) * (1.0f / FY_) * d : qnan;
    Z[i] = valid ? d : qnan;
    // pads beyond N: -inf so they can never win the argmax
    dist[i] = inb ? inf : -inf;
  }

  u32* ctrl = (u32*)ws;                    // [arrival count, epoch]
  u64* slots = ws + 1;                     // per-iteration argmax slots

  float px, py, pz;
  point_from_index(depth, 0, px, py, pz);  // first FPS point = index 0

  for (int k = 1; k < M_PTS; ++k) {
    float bestd = -inf;
    int besti = 0;
#pragma unroll
    for (int i = 0; i < PPT; ++i) {
      float dx = X[i] - px, dy = Y[i] - py, dz = Z[i] - pz;
      float c = fmaf(dz, dz, fmaf(dy, dy, dx * dx));
      float nd = fminf(dist[i], c);
      dist[i] = nd;
      if (nd > bestd) { bestd = nd; besti = base + i * TOT; }
    }
    // pack (ordered key | ~idx) so max => largest dist, ties => smallest index
    u64 key = ((u64)fkey(bestd) << 32) | (u64)(u32)(~(u32)besti);
    key = wave_max_u64(key);
    const int wid = t >> 5, lid = t & 31;
    if (lid == 0) sred[wid] = key;
    __syncthreads();
    if (wid == 0) {
      u64 v = sred[lid];                   // 32 waves -> one full wave32 reduce
      v = wave_max_u64(v);
      if (lid == 0) atomicMax(&slots[k], v);
    }
    grid_barrier(ctrl, ctrl + 1, (u32)k);
    u64 win = __hip_atomic_load(&slots[k], __ATOMIC_ACQUIRE, __HIP_MEMORY_SCOPE_AGENT);
    int widx = (int)(~(u32)win);
    point_from_index(depth, widx, px, py, pz);
  }
}

// ---------------- epilogue: gather + normalize + write [2048 x 9] ----------------
__device__ __forceinline__ float blk_red(float v, bool domax) {
  __shared__ float s[32];
#pragma unroll
  for (int off = 16; off > 0; off >>= 1) {
    float o = __shfl_xor(v, off, 32);
    v = domax ? fmaxf(v, o) : fminf(v, o);
  }
  int wid = threadIdx.x >> 5, lid = threadIdx.x & 31;
  if (lid == 0) s[wid] = v;
  __syncthreads();
  if (threadIdx.x == 0) {
    float m = s[0];
    for (int i = 1; i < 32; ++i) m = domax ? fmaxf(m, s[i]) : fminf(m, s[i]);
    s[0] = m;
  }
  __syncthreads();
  float r = s[0];
  __syncthreads();
  return r;
}

__global__ __launch_bounds__(1024) void epilogue_kernel(const float* __restrict__ depth,
                                                        const float* __restrict__ rgb,
                                                        const u64* __restrict__ ws,
                                                        float* __restrict__ out) {
  const u64* slots = ws + 1;
  const float inf = __builtin_inff();
  float xs[2], ys[2], zs[2];
#pragma unroll
  for (int e = 0; e < 2; ++e) {
    int j = (int)threadIdx.x + e * 1024;
    int widx = (j == 0) ? 0 : (int)(~(u32)slots[j]);
    float d = depth[widx];
    bool valid = (d > MIN_D) && (d < MAX_D);
    int u = widx % W_IMG;
    int v = widx / W_IMG;
    float x = valid ? ((float)u - CX_) * d * (1.0f / FX_) : inf;
    float y = valid ? ((float)v - CY_) * d * (1.0f / FY_) : inf;
    float z = valid ? d : inf;
    float r = valid ? rgb[widx * 3 + 0] * (1.0f / 255.0f) : 0.0f;
    float g = valid ? rgb[widx * 3 + 1] * (1.0f / 255.0f) : 0.0f;
    float bl = valid ? rgb[widx * 3 + 2] * (1.0f / 255.0f) : 0.0f;
    out[j * 9 + 0] = x; out[j * 9 + 1] = y; out[j * 9 + 2] = z;
    out[j * 9 + 3] = r; out[j * 9 + 4] = g; out[j * 9 + 5] = bl;
    xs[e] = x; ys[e] = y; zs[e] = z;
  }
  float mnx = blk_red(fminf(xs[0], xs[1]), false);
  float mny = blk_red(fminf(ys[0], ys[1]), false);
  float mnz = blk_red(fminf(zs[0], zs[1]), false);
  float cx0 = xs[0] - mnx, cx1 = xs[1] - mnx;
  float cy0 = ys[0] - mny, cy1 = ys[1] - mny;
  float cz0 = zs[0] - mnz, cz1 = zs[1] - mnz;
  float mxx = blk_red(fmaxf(cx0, cx1), true);
  float mxy = blk_red(fmaxf(cy0, cy1), true);
  float mxz = blk_red(fmaxf(cz0, cz1), true);
  mxx = (mxx < 1e-8f) ? 1.0f : mxx;
  mxy = (mxy < 1e-8f) ? 1.0f : mxy;
  mxz = (mxz < 1e-8f) ? 1.0f : mxz;
#pragma unroll
  for (int e = 0; e < 2; ++e) {
    int j = (int)threadIdx.x + e * 1024;
    out[j * 9 + 6] = (e ? cx1 : cx0) / mxx;
    out[j * 9 + 7] = (e ? cy1 : cy0) / mxy;
    out[j * 9 + 8] = (e ? cz1 : cz0) / mxz;
  }
}

extern "C" void kernel_launch(void* const* d_in, const int* in_sizes, int n_in,
                              void* d_out, int out_size, void* d_ws, size_t ws_size,
                              hipStream_t stream) {
  (void)in_sizes; (void)n_in; (void)out_size; (void)ws_size;
  const float* depth = (const float*)d_in[0];
  const float* rgb   = (const float*)d_in[1];
  float* out = (float*)d_out;
  u64* ws = (u64*)d_ws;

  init_ws_kernel<<<(WS_U64 + 255) / 256, 256, 0, stream>>>(ws);
  fps_kernel<<<NB, NT, 0, stream>>>(depth, ws);
  epilogue_kernel<<<1, 1024, 0, stream>>>(depth, rgb, ws, out);
}